// GaussianCopulaTransform_85521388798272
// MI455X (gfx1250) — compile-verified
//
#include <hip/hip_runtime.h>

typedef __attribute__((ext_vector_type(2))) float v2f;
typedef __attribute__((ext_vector_type(8))) float v8f;

#define B_ 32
#define L_ 4096
#define D_ 256

// Giles' single-precision erfinv approximation
__device__ __forceinline__ float erfinv_f(float y) {
    float w = -__logf((1.0f - y) * (1.0f + y));
    float p;
    if (w < 5.0f) {
        w -= 2.5f;
        p = 2.81022636e-08f;
        p = fmaf(p, w, 3.43273939e-07f);
        p = fmaf(p, w, -3.5233877e-06f);
        p = fmaf(p, w, -4.39150654e-06f);
        p = fmaf(p, w, 0.00021858087f);
        p = fmaf(p, w, -0.00125372503f);
        p = fmaf(p, w, -0.00417768164f);
        p = fmaf(p, w, 0.246640727f);
        p = fmaf(p, w, 1.50140941f);
    } else {
        w = sqrtf(w) - 3.0f;
        p = -0.000200214257f;
        p = fmaf(p, w, 0.000100950558f);
        p = fmaf(p, w, 0.00134934322f);
        p = fmaf(p, w, -0.00367342844f);
        p = fmaf(p, w, 0.00573950773f);
        p = fmaf(p, w, -0.0076224613f);
        p = fmaf(p, w, 0.00943887047f);
        p = fmaf(p, w, 1.00167406f);
        p = fmaf(p, w, 2.83297682f);
    }
    return p * y;
}

// One workgroup per (b, d) column: bitonic sort -> ranks -> z = Phi^-1(u),
// plus per-column mean into workspace.
__global__ __launch_bounds__(256) void rank_z_kernel(const float* __restrict__ x,
                                                     float* __restrict__ z,
                                                     float* __restrict__ mean) {
    __shared__ float skey[L_];
    __shared__ int   sidx[L_];
    const int col = blockIdx.x;        // 0 .. B*D-1
    const int b   = col >> 8;          // col / D_
    const int d   = col & (D_ - 1);
    const int tid = threadIdx.x;
    const size_t base = (size_t)b * L_ * D_ + (size_t)d;

    for (int i = tid; i < L_; i += 256) {
        skey[i] = x[base + (size_t)i * D_];
        sidx[i] = i;
    }
    __syncthreads();

    // Bitonic sort ascending on (key, idx)
    for (unsigned k = 2; k <= (unsigned)L_; k <<= 1) {
        for (unsigned j = k >> 1; j > 0; j >>= 1) {
            for (unsigned i = tid; i < (unsigned)L_; i += 256) {
                unsigned ixj = i ^ j;
                if (ixj > i) {
                    bool up = ((i & k) == 0);
                    float a = skey[i], c = skey[ixj];
                    if (up ? (a > c) : (a < c)) {
                        skey[i] = c; skey[ixj] = a;
                        int t = sidx[i]; sidx[i] = sidx[ixj]; sidx[ixj] = t;
                    }
                }
            }
            __syncthreads();
        }
    }

    // rank of the element at sorted position p is p+1 (distinct values)
    const float invL = 1.0f / (float)L_;
    float lsum = 0.0f;
    for (int p = tid; p < L_; p += 256) {
        int orig = sidx[p];
        float u = ((float)(p + 1) + 0.5f) * invL;
        u = fminf(fmaxf(u, 1e-6f), 1.0f - 1e-6f);
        float zz = 1.41421356237f * erfinv_f(2.0f * u - 1.0f);
        z[base + (size_t)orig * D_] = zz;
        lsum += zz;
    }
    __syncthreads();
    skey[tid] = lsum;             // reuse LDS for block reduction
    __syncthreads();
    for (int s = 128; s > 0; s >>= 1) {
        if (tid < s) skey[tid] += skey[tid + s];
        __syncthreads();
    }
    if (tid == 0) mean[col] = skey[0] * invL;
}

// cov[b] = Zc^T * Zc / (L-1) via V_WMMA_F32_16X16X4_F32.
// grid = (B, 4); 8 waves/block -> 32 waves per batch.
// Each wave owns a 4x2 block of 16x16 tiles (64x32 output region):
// 12 fragment loads feed 8 WMMAs per K=4 step (1.5 loads/WMMA), and the
// 8 independent accumulators break the WMMA->WMMA RAW chain on the XDL pipe.
__global__ __launch_bounds__(256) void cov_wmma_kernel(const float* __restrict__ z,
                                                       const float* __restrict__ mean,
                                                       float* __restrict__ cov) {
    const int b    = blockIdx.x;
    const int wv   = blockIdx.y * 8 + (threadIdx.x >> 5);  // 0..31
    const int lane = threadIdx.x & 31;
    const int ti0  = (wv >> 3) * 4;   // tile-row group: 4 tiles tall
    const int tj0  = (wv & 7) * 2;    // tile-col group: 2 tiles wide

    const int mn = lane & 15;                 // M (for A) / N (for B)
    const int kb = (lane >> 4) << 1;          // lanes 0-15: K{0,1}; lanes 16-31: K{2,3}

    const float* zb = z + (size_t)b * L_ * D_;

    const float* pA[4];
    const float* pB[2];
    float ma[4], mb[2];
    #pragma unroll
    for (int t = 0; t < 4; ++t) {
        int da = (ti0 + t) * 16 + mn;
        pA[t] = zb + da;
        ma[t] = mean[b * D_ + da];
    }
    #pragma unroll
    for (int t = 0; t < 2; ++t) {
        int db = (tj0 + t) * 16 + mn;
        pB[t] = zb + db;
        mb[t] = mean[b * D_ + db];
    }

    v8f acc[8];
    #pragma unroll
    for (int i = 0; i < 8; ++i) acc[i] = (v8f){};

    for (int k = 0; k < L_; k += 4) {
        const size_t r0 = (size_t)(k + kb) * D_;
        const size_t r1 = r0 + D_;
        v2f af[4], bf[2];
        #pragma unroll
        for (int t = 0; t < 4; ++t) {
            af[t].x = pA[t][r0] - ma[t];
            af[t].y = pA[t][r1] - ma[t];
        }
        #pragma unroll
        for (int t = 0; t < 2; ++t) {
            bf[t].x = pB[t][r0] - mb[t];
            bf[t].y = pB[t][r1] - mb[t];
        }
        #pragma unroll
        for (int i = 0; i < 4; ++i) {
            #pragma unroll
            for (int j = 0; j < 2; ++j) {
                acc[i * 2 + j] = __builtin_amdgcn_wmma_f32_16x16x4_f32(
                    false, af[i], false, bf[j], (short)0, acc[i * 2 + j], false, false);
            }
        }
    }

    const float s = 1.0f / (float)(L_ - 1);
    // C/D layout: VGPR v -> lanes 0-15: M=v, N=lane; lanes 16-31: M=v+8, N=lane-16
    const int rsel = (lane >> 4) << 3;
    const int cn   = lane & 15;
    float* cb = cov + ((size_t)b * D_ * D_);
    #pragma unroll
    for (int i = 0; i < 4; ++i) {
        #pragma unroll
        for (int j = 0; j < 2; ++j) {
            const int row0 = (ti0 + i) * 16 + rsel;
            const int col0 = (tj0 + j) * 16 + cn;
            #pragma unroll
            for (int v = 0; v < 8; ++v) {
                cb[(size_t)(row0 + v) * D_ + col0] = acc[i * 2 + j][v] * s;
            }
        }
    }
}

// corr = cov / (std_i * std_j + 1e-8)
__global__ __launch_bounds__(256) void corr_kernel(const float* __restrict__ cov,
                                                   float* __restrict__ corr) {
    int i = blockIdx.x * 256 + threadIdx.x;   // grid covers exactly B*D*D
    int b  = i >> 16;                         // D*D = 65536
    int rc = i & 65535;
    int r = rc >> 8, c = rc & (D_ - 1);
    const float* cb = cov + ((size_t)b << 16);
    float cij = cb[rc];
    float si = sqrtf(cb[r * D_ + r]);
    float sj = sqrtf(cb[c * D_ + c]);
    corr[i] = cij / (si * sj + 1e-8f);
}

extern "C" void kernel_launch(void* const* d_in, const int* in_sizes, int n_in,
                              void* d_out, int out_size, void* d_ws, size_t ws_size,
                              hipStream_t stream) {
    (void)in_sizes; (void)n_in; (void)out_size; (void)ws_size;
    const float* x = (const float*)d_in[0];
    float* z    = (float*)d_out;                               // [B, L, D]
    float* corr = (float*)d_out + (size_t)B_ * L_ * D_;        // [B, D, D]
    float* mean = (float*)d_ws;                                // [B*D]
    float* cov  = (float*)d_ws + (size_t)(B_ * D_);            // [B, D, D]

    rank_z_kernel<<<dim3(B_ * D_), 256, 0, stream>>>(x, z, mean);
    cov_wmma_kernel<<<dim3(B_, 4), 256, 0, stream>>>(z, mean, cov);
    corr_kernel<<<dim3((B_ * D_ * D_) / 256), 256, 0, stream>>>(cov, corr);
}